// RGCN_8375186227325
// MI455X (gfx1250) — compile-verified
//
#include <hip/hip_runtime.h>

// Problem constants (fixed by the reference harness)
#define NNODES 50000
#define NEDGES 600000
#define HDIM   128
#define OUTDIM 32
#define NREL   64
#define NBASE  16
#define HC     32               // feature-chunk width == wave32 lanes
#define KTOT   (NBASE * HC)     // 512: contraction length of each chunk-GEMM

typedef float v2f __attribute__((ext_vector_type(2)));
typedef float v8f __attribute__((ext_vector_type(8)));

// ---------------------------------------------------------------- zero agg
__global__ __launch_bounds__(256) void rgcn_zero_kernel(float4* __restrict__ p, int n4) {
    int i = blockIdx.x * 256 + threadIdx.x;
    if (i < n4) p[i] = make_float4(0.f, 0.f, 0.f, 0.f);
}

// ------------------------------------------------- edge scatter (atomics)
// agg[d, b, lane] += comp[etype, b] * norm * x[src, chunkBase + lane]
// One wave per edge; lane = feature within chunk; comp staged in LDS.
// Edge scalars go through readfirstlane so they become s_loads + SGPR base.
// Atomics are hand-emitted global_atomic_add_f32 (device scope, L2 atomic
// units) with literal immediate offsets off one shared address VGPR pair.
__global__ __launch_bounds__(256) void rgcn_scatter_kernel(
    const float* __restrict__ x,     // [N, HDIM]
    const int*   __restrict__ src,
    const int*   __restrict__ dst,
    const int*   __restrict__ et,
    const float* __restrict__ norm,  // [E]
    const float* __restrict__ comp,  // [NREL, NBASE]
    float*       __restrict__ agg,   // [N, NBASE, HC]
    int chunkBase)
{
    __shared__ float scomp[NREL * NBASE];   // 4 KB
    for (int i = threadIdx.x; i < NREL * NBASE; i += 256) scomp[i] = comp[i];
    __syncthreads();

    int lane = threadIdx.x & 31;
    int e    = blockIdx.x * 8 + (threadIdx.x >> 5);
    if (e >= NEDGES) return;
    e = __builtin_amdgcn_readfirstlane(e);      // wave-uniform -> scalar loads

    int   s  = src[e];
    int   d  = dst[e];
    int   r  = et[e];
    float nm = norm[e];
    float xv = x[(size_t)s * HDIM + chunkBase + lane] * nm;

    const float* c = scomp + r * NBASE;
    float cv[NBASE];
#pragma unroll
    for (int b = 0; b < NBASE; ++b) cv[b] = c[b];

    float* arow = agg + (size_t)d * KTOT + lane;
#define RGCN_AT(B, OFF)                                                       \
    asm volatile("global_atomic_add_f32 %0, %1, off offset:" OFF              \
                 " scope:SCOPE_DEV"                                           \
                 :: "v"(arow), "v"(cv[B] * xv) : "memory")
    RGCN_AT(0,  "0");    RGCN_AT(1,  "128");  RGCN_AT(2,  "256");  RGCN_AT(3,  "384");
    RGCN_AT(4,  "512");  RGCN_AT(5,  "640");  RGCN_AT(6,  "768");  RGCN_AT(7,  "896");
    RGCN_AT(8,  "1024"); RGCN_AT(9,  "1152"); RGCN_AT(10, "1280"); RGCN_AT(11, "1408");
    RGCN_AT(12, "1536"); RGCN_AT(13, "1664"); RGCN_AT(14, "1792"); RGCN_AT(15, "1920");
#undef RGCN_AT
}

// -------------------------------------------- fp32 WMMA GEMM per chunk
// out[v, o] (+)= sum_{k=(b,il)} agg[v, k] * V[b, chunkBase+il, o]  (+bias / relu)
// One wave computes a 16x16 output tile; K loop in steps of 4 via
// V_WMMA_F32_16X16X4_F32 (full fp32 precision, matches fp32 reference).
// ACC/RELU are compile-time so the epilogue is branch-free and its loads clause.
template <int OD, bool ACC, bool RELU>
__global__ __launch_bounds__(256) void rgcn_wmma_gemm_kernel(
    const float* __restrict__ agg,   // [N, KTOT]
    const float* __restrict__ V,     // [NBASE, HDIM, OD]
    const float* __restrict__ bias,  // [OD]
    float*       __restrict__ out,   // [N, OD]
    int chunkBase)
{
    const int colTiles = OD / 16;
    int wave    = (blockIdx.x * 256 + threadIdx.x) >> 5;
    int lane    = threadIdx.x & 31;
    int rowTile = wave / colTiles;
    int colTile = wave - rowTile * colTiles;
    if (rowTile >= NNODES / 16) return;    // whole-wave exit: EXEC stays all-1s for WMMA

    int m  = lane & 15;       // row within A / col within B,C,D
    int hi = lane >> 4;       // lane group: K offset {0,1} vs {2,3}; D row +8

    const float* arow = agg + (size_t)(rowTile * 16 + m) * KTOT;
    int          col  = colTile * 16 + m;

    v8f acc = {};
    for (int k = 0; k < KTOT; k += 4) {
        int kk = k + hi * 2;                       // even, never crosses a basis boundary
        // A frag: lane holds A[m][kk], A[m][kk+1] -> contiguous float2
        v2f a = *(const v2f*)(arow + kk);
        // B frag: lane holds B[kk][col], B[kk+1][col]
        int b_  = kk >> 5;                         // basis index (HC == 32)
        int il  = kk & 31;                         // feature within chunk
        const float* vb = V + ((size_t)b_ * HDIM + chunkBase + il) * OD + col;
        v2f bv;
        bv.x = vb[0];
        bv.y = vb[OD];
        acc = __builtin_amdgcn_wmma_f32_16x16x4_f32(
            /*neg_a=*/false, a, /*neg_b=*/false, bv,
            /*c_mod=*/(short)0, acc, /*reuse_a=*/false, /*reuse_b=*/false);
    }

    // ---- epilogue (compile-time specialized) ----
    float* obase = out + (size_t)(rowTile * 16 + hi * 8) * OD + col;
    float  prev[8];
    if (ACC) {
#pragma unroll
        for (int j = 0; j < 8; ++j)                // clause of 8 loads, one wait
            prev[j] = obase[(size_t)j * OD];
    } else {
        float b = bias[col];
#pragma unroll
        for (int j = 0; j < 8; ++j) prev[j] = b;
    }
#pragma unroll
    for (int j = 0; j < 8; ++j) {
        float v = acc[j] + prev[j];
        if (RELU) v = fmaxf(v, 0.f);
        obase[(size_t)j * OD] = v;
    }
}

// ---------------------------------------------------------------- driver
extern "C" void kernel_launch(void* const* d_in, const int* in_sizes, int n_in,
                              void* d_out, int out_size, void* d_ws, size_t ws_size,
                              hipStream_t stream)
{
    const int*   src   = (const int*)  d_in[0];
    const int*   dst   = (const int*)  d_in[1];
    const int*   et    = (const int*)  d_in[2];
    const float* norm  = (const float*)d_in[3];
    const float* emb   = (const float*)d_in[4];
    const float* V1    = (const float*)d_in[5];
    const float* comp1 = (const float*)d_in[6];
    const float* bias1 = (const float*)d_in[7];
    const float* V2    = (const float*)d_in[8];
    const float* comp2 = (const float*)d_in[9];
    const float* bias2 = (const float*)d_in[10];
    float*       out   = (float*)d_out;

    // Workspace: agg chunk [N, NBASE, HC] f32 = 102.4 MB (L2-resident target),
    //            h1 [N, HDIM] f32 = 25.6 MB.
    float* agg = (float*)d_ws;
    float* h1  = (float*)((char*)d_ws + (size_t)NNODES * KTOT * sizeof(float));

    const int aggN4 = NNODES * KTOT / 4;                       // 6.4M float4
    dim3 blk(256);
    dim3 zgrid((aggN4 + 255) / 256);                           // 25000
    dim3 sgrid(NEDGES / 8);                                    // 75000 (exact)
    dim3 g1grid((NNODES / 16) * (HDIM / 16) / 8);              // 3125 (exact)
    dim3 g2grid(((NNODES / 16) * (OUTDIM / 16) + 7) / 8);      // 782

    const int NCHUNK = HDIM / HC;   // 4

    // ---- layer 1: emb -> h1 (relu on last chunk)
    for (int c = 0; c < NCHUNK; ++c) {
        rgcn_zero_kernel<<<zgrid, blk, 0, stream>>>((float4*)agg, aggN4);
        rgcn_scatter_kernel<<<sgrid, blk, 0, stream>>>(emb, src, dst, et, norm,
                                                       comp1, agg, c * HC);
        if (c == 0)
            rgcn_wmma_gemm_kernel<HDIM, false, false><<<g1grid, blk, 0, stream>>>(
                agg, V1, bias1, h1, c * HC);
        else if (c < NCHUNK - 1)
            rgcn_wmma_gemm_kernel<HDIM, true, false><<<g1grid, blk, 0, stream>>>(
                agg, V1, bias1, h1, c * HC);
        else
            rgcn_wmma_gemm_kernel<HDIM, true, true><<<g1grid, blk, 0, stream>>>(
                agg, V1, bias1, h1, c * HC);
    }
    // ---- layer 2: h1 -> out
    for (int c = 0; c < NCHUNK; ++c) {
        rgcn_zero_kernel<<<zgrid, blk, 0, stream>>>((float4*)agg, aggN4);
        rgcn_scatter_kernel<<<sgrid, blk, 0, stream>>>(h1, src, dst, et, norm,
                                                       comp2, agg, c * HC);
        if (c == 0)
            rgcn_wmma_gemm_kernel<OUTDIM, false, false><<<g2grid, blk, 0, stream>>>(
                agg, V2, bias2, out, c * HC);
        else
            rgcn_wmma_gemm_kernel<OUTDIM, true, false><<<g2grid, blk, 0, stream>>>(
                agg, V2, bias2, out, c * HC);
    }
}